// LabelLoss_33234456937090
// MI455X (gfx1250) — compile-verified
//
#include <hip/hip_runtime.h>

typedef float v2f __attribute__((ext_vector_type(2)));
typedef float v4f __attribute__((ext_vector_type(4)));
typedef float v8f __attribute__((ext_vector_type(8)));

#define HWSZ 65536   // 256*256
#define NB   32
#define NJ   16
#define NC   7

// ---------------------------------------------------------------------------
// Kernel 1: streaming argmax over heatmap[b,j,:,:] (the 128 MiB, HBM-bound part)
// One block per (b,j); 256 threads x 64 float4 NT loads each; LDS tree reduce
// with lowest-flat-index tie-break to match jnp.argmax semantics.
// ---------------------------------------------------------------------------
__global__ __launch_bounds__(256) void hm_argmax_kernel(
    const float* __restrict__ heatmap,
    float* __restrict__ wsVal,
    int*   __restrict__ wsIdx) {
  __shared__ float sVal[256];
  __shared__ int   sIdx[256];

  const int bj  = blockIdx.x;          // 0..511 -> (b*16 + j)
  const int tid = threadIdx.x;
  const v4f* base = (const v4f*)(heatmap + (size_t)bj * HWSZ);

  float best    = -INFINITY;
  int   bestIdx = 0x7fffffff;

  // 65536 floats = 16384 float4; stride-256 coalesced, non-temporal (read-once)
  #pragma unroll 4
  for (int i = tid; i < HWSZ / 4; i += 256) {
    v4f v = __builtin_nontemporal_load(&base[i]);
    float m01 = fmaxf(v.x, v.y);
    float m23 = fmaxf(v.z, v.w);
    float m   = fmaxf(m01, m23);
    if (m > best) {                    // strict '>' keeps earliest index in-thread
      best = m;
      const int off = i * 4;
      bestIdx = (v.x == m) ? off
              : (v.y == m) ? off + 1
              : (v.z == m) ? off + 2
              :              off + 3;
    }
  }

  sVal[tid] = best;
  sIdx[tid] = bestIdx;
  __syncthreads();

  for (int s = 128; s > 0; s >>= 1) {
    if (tid < s) {
      const float ov = sVal[tid + s];
      const int   oi = sIdx[tid + s];
      if (ov > sVal[tid] || (ov == sVal[tid] && oi < sIdx[tid])) {
        sVal[tid] = ov;
        sIdx[tid] = oi;
      }
    }
    __syncthreads();
  }

  if (tid == 0) {
    wsVal[bj] = sVal[0];
    wsIdx[bj] = sIdx[0];
  }
}

// ---------------------------------------------------------------------------
// Kernel 2: per-batch loss via WMMA.
// D[j][c] = mask_j ? (pred[b,c,peak] - gt[b,j,c]) : 0, zero-padded to 16x8.
// class_loss[j] = (D * D^T)[j][j]; out[b] = sum_j class_loss[j].
// f32 16x16x4 WMMA: A-layout of D and B-layout of D^T have identical register
// images, so the same v2f feeds SRC0 and SRC1. Two WMMAs chain K=0..3, K=4..7.
// ---------------------------------------------------------------------------
__global__ __launch_bounds__(32) void loss_wmma_kernel(
    const float* __restrict__ pred,
    const float* __restrict__ gt,
    const float* __restrict__ wsVal,
    const int*   __restrict__ wsIdx,
    float* __restrict__ out) {
  __shared__ float Ds[NJ][8];

  const int b    = blockIdx.x;
  const int lane = threadIdx.x;        // wave32: one wave per block

  if (lane < NJ) {
    const int   idx  = wsIdx[b * NJ + lane];
    const float peak = wsVal[b * NJ + lane];
    const bool  mask = (peak == 1.0f);
    const float* pb = pred + (size_t)b * NC * HWSZ + idx;   // pred[b,c,x,y]
    const float* gb = gt + ((size_t)b * NJ + lane) * NC;
    #pragma unroll
    for (int c = 0; c < NC; ++c) {
      const float d = pb[(size_t)c * HWSZ] - gb[c];
      Ds[lane][c] = mask ? d : 0.0f;
    }
    Ds[lane][7] = 0.0f;                // pad K to 8
  }
  __syncthreads();

  // A (16x4 f32) layout: lanes 0-15 = row m, {K0,K1}; lanes 16-31 = row m, {K2,K3}
  const int m    = lane & 15;
  const int koff = (lane >> 4) * 2;
  v2f a0, a1;
  a0.x = Ds[m][koff];
  a0.y = Ds[m][koff + 1];
  a1.x = Ds[m][4 + koff];
  a1.y = Ds[m][4 + koff + 1];

  v8f acc = {};
  // D = A x B + C ; args: (neg_a, A, neg_b, B, c_mod, C, reuse_a, reuse_b)
  acc = __builtin_amdgcn_wmma_f32_16x16x4_f32(false, a0, false, a0, (short)0,
                                              acc, false, false);
  acc = __builtin_amdgcn_wmma_f32_16x16x4_f32(false, a1, false, a1, (short)0,
                                              acc, false, false);

  // Diagonal of 16x16 f32 C/D: (j,j) lives at VGPR j, lane j   (j < 8)
  //                            and VGPR j-8, lane j+16         (j >= 8)
  const int sel = (lane < 8) ? lane : ((lane >= 24) ? (lane - 24) : -1);
  float contrib = 0.0f;
  #pragma unroll
  for (int r = 0; r < 8; ++r) contrib = (sel == r) ? acc[r] : contrib;

  #pragma unroll
  for (int off = 16; off > 0; off >>= 1)
    contrib += __shfl_xor(contrib, off, 32);

  if (lane == 0) out[b] = contrib;
}

// ---------------------------------------------------------------------------
extern "C" void kernel_launch(void* const* d_in, const int* in_sizes, int n_in,
                              void* d_out, int out_size, void* d_ws, size_t ws_size,
                              hipStream_t stream) {
  const float* pred    = (const float*)d_in[0];  // (32,7,256,256)
  const float* gt      = (const float*)d_in[1];  // (32,16,7)
  const float* heatmap = (const float*)d_in[2];  // (32,16,256,256)
  float* out = (float*)d_out;                    // (32,)

  float* wsVal = (float*)d_ws;                              // 512 floats
  int*   wsIdx = (int*)((char*)d_ws + NB * NJ * sizeof(float)); // 512 ints

  hm_argmax_kernel<<<NB * NJ, 256, 0, stream>>>(heatmap, wsVal, wsIdx);
  loss_wmma_kernel<<<NB, 32, 0, stream>>>(pred, gt, wsVal, wsIdx, out);
}